// GNN_Critic_35450660061514
// MI455X (gfx1250) — compile-verified
//
#include <hip/hip_runtime.h>
#include <math.h>

// GCN critic for MI455X (gfx1250, wave32).
// - Big GEMMs (h1@W2: 100000x128x128, g@Wm1: 1000x128x128) use
//   V_WMMA_F32_16X16X4_F32 (full f32; the problem is L2/scatter bound, so
//   low-precision WMMA buys nothing and would hurt accuracy).
// - Weights are pre-repacked into K-pair-interleaved layout so each B
//   fragment is one aligned ds_load_b64 (no VALU repack in the WMMA loop).
// - Weight staging into LDS uses the Tensor Data Mover
//   (tensor_load_to_lds + s_wait_tensorcnt) issued by wave 0; an asm
//   memory-clobber makes its LDS writes visible to the compiler so the
//   ds reads are not folded away.
// - Working set (~103MB) fits in the 192MB L2, so edge gather/scatter
//   atomics run out of L2, not HBM.

typedef __attribute__((ext_vector_type(2))) float v2f;
typedef __attribute__((ext_vector_type(8))) float v8f;
typedef __attribute__((ext_vector_type(4))) unsigned int u32x4;
typedef __attribute__((ext_vector_type(8))) int i32x8;
typedef __attribute__((ext_vector_type(4))) int i32x4;

#define HID 128

__global__ void k_fill1(float* p, int n) {
  int i = blockIdx.x * blockDim.x + threadIdx.x;
  if (i < n) p[i] = 1.0f;
}

__global__ void k_zero(float* p, int n) {
  int i = blockIdx.x * blockDim.x + threadIdx.x;
  if (i < n) p[i] = 0.0f;
}

__global__ void k_degacc(const int* __restrict__ dst, float* deg, int E) {
  int i = blockIdx.x * blockDim.x + threadIdx.x;
  if (i < E) atomicAdd(&deg[dst[i]], 1.0f);
}

__global__ void k_rsqrt(float* p, int n) {
  int i = blockIdx.x * blockDim.x + threadIdx.x;
  if (i < n) p[i] = rsqrtf(p[i]);
}

// Repack 128x128 row-major W into K-pair-interleaved layout:
// Bp[(k/2)*256 + c*2 + (k&1)] = W[k*128 + c]
// so a B WMMA fragment (W[kk][c], W[kk+1][c]) is one contiguous b64.
__global__ void k_repack(const float* __restrict__ B, float* __restrict__ Bp) {
  int i = blockIdx.x * blockDim.x + threadIdx.x;
  if (i >= HID * HID) return;
  int k = i >> 7, c = i & 127;
  Bp[(k >> 1) * 256 + c * 2 + (k & 1)] = B[i];
}

// T[n,f] = sum_k x[n,k] * W1[k,f]   (K=10, skinny: plain VALU)
__global__ void k_xw1(const float* __restrict__ x, const float* __restrict__ W1,
                      float* __restrict__ T, int N) {
  int i = blockIdx.x * blockDim.x + threadIdx.x;
  if (i >= N * HID) return;
  int n = i >> 7, f = i & 127;
  const float* xr = x + n * 10;
  float s = 0.f;
#pragma unroll
  for (int k = 0; k < 10; ++k) s = fmaf(xr[k], W1[k * HID + f], s);
  T[i] = s;
}

// AGG[n,:] = T[n,:] * dinv[n]^2   (self-loop contribution, avoids N extra atomics)
__global__ void k_selfinit(const float* __restrict__ T, const float* __restrict__ dinv,
                           float* __restrict__ AGG, int N) {
  int i = blockIdx.x * blockDim.x + threadIdx.x;
  if (i >= N * HID) return;
  float d = dinv[i >> 7];
  AGG[i] = T[i] * d * d;
}

// AGG[dst,:] += T[src,:] * dinv[src]*dinv[dst], one thread per (edge, channel)
__global__ void k_scatter(const int* __restrict__ src, const int* __restrict__ dst,
                          const float* __restrict__ dinv, const float* __restrict__ T,
                          float* __restrict__ AGG, int E) {
  long long gid = (long long)blockIdx.x * blockDim.x + threadIdx.x;
  int e = (int)(gid >> 7);
  if (e >= E) return;
  int f = (int)(gid & 127);
  int s = src[e], d = dst[e];
  float norm = dinv[s] * dinv[d];
  atomicAdd(&AGG[d * HID + f], T[s * HID + f] * norm);
}

__global__ void k_biasrelu(const float* __restrict__ A, const float* __restrict__ b,
                           float* __restrict__ H, int N) {
  int i = blockIdx.x * blockDim.x + threadIdx.x;
  if (i >= N * HID) return;
  H[i] = fmaxf(A[i] + b[i & 127], 0.f);
}

// K-loop over 128 with 16x16x4 f32 WMMA; TAIL version masks out-of-range rows.
template <bool TAIL>
__device__ __forceinline__ void kloop128(const float* __restrict__ Arow,
                                         const float* __restrict__ Bs,
                                         float msk, int khalf, int rlo, v8f acc[8]) {
  for (int k0 = 0; k0 < HID; k0 += 4) {
    int kk = k0 + khalf * 2;          // A layout: lanes16-31 hold K pair {2,3}
    v2f a;
    a.x = Arow[kk];
    a.y = Arow[kk + 1];
    if (TAIL) { a.x *= msk; a.y *= msk; }
    const float* bp = Bs + ((k0 >> 1) + khalf) * 256 + rlo * 2;  // paired layout
#pragma unroll
    for (int t = 0; t < 8; ++t) {
      v2f b = *(const v2f*)(bp + t * 32);   // one ds_load_b64 per fragment
      acc[t] = __builtin_amdgcn_wmma_f32_16x16x4_f32(
          /*neg_a=*/false, a, /*neg_b=*/false, b,
          /*c_mod=*/(short)0, acc[t], /*reuse_a=*/false, /*reuse_b=*/false);
    }
  }
}

// C[M,128] = A[M,128] @ Bp (K-pair-repacked 128x128) (+bias, +tanh).
// Bp staged into LDS by the Tensor Data Mover; one wave per 16-row M tile.
__global__ void __launch_bounds__(256)
k_gemm128_wmma(const float* __restrict__ A, const float* __restrict__ Bp,
               float* __restrict__ C, int M, const float* __restrict__ bias, int act) {
  __shared__ float Bs[HID * HID];   // 64KB; only LDS object -> offset 0

#if __has_builtin(__builtin_amdgcn_tensor_load_to_lds)
  if (threadIdx.x < 32) {           // wave 0 drives the TDM for the workgroup
    unsigned long long ga = (unsigned long long)(uintptr_t)Bp;
    u32x4 g0;
    g0[0] = 1u;                                        // count=1 (valid user D#)
    g0[1] = 0u;                                        // lds_addr = 0 (Bs)
    g0[2] = (unsigned)(ga & 0xFFFFFFFFu);              // global_addr[31:0]
    g0[3] = (unsigned)((ga >> 32) & 0x01FFFFFFu)       // global_addr[56:32]
            | 0x80000000u;                             // type=2 ("image")
    i32x8 g1;
    g1[0] = 0x00020000;          // workgroup_mask=0, data_size=2 (4 bytes)
    g1[1] = (int)(128u << 16);   // tensor_dim0 = 128
    g1[2] = (int)(128u << 16);   // tensor_dim1 = 128
    g1[3] = (int)(128u << 16);   // tile_dim0 = 128
    g1[4] = 128;                 // tile_dim1 = 128, tile_dim2 = 0
    g1[5] = 128;                 // tensor_dim0_stride = 128
    g1[6] = 0;
    g1[7] = 0;
    i32x4 z4 = {0, 0, 0, 0};              // groups 2/3 unused (2D tensor)
    i32x8 z8 = {0, 0, 0, 0, 0, 0, 0, 0};  // extra group (clang-23 6-arg form)
    __builtin_amdgcn_tensor_load_to_lds(g0, g1, z4, z4, z8, 0);
    __builtin_amdgcn_s_wait_tensorcnt((short)0);
  }
  // The TDM builtin's side effects don't alias Bs in the compiler's view,
  // so without this the ds reads of Bs get folded to undef (seen in round-3
  // disasm: constant B operand, ds count ~0). Escape Bs's address through an
  // asm with a full memory clobber so Bs is treated as written.
  {
    void* escaped = (void*)Bs;
    asm volatile("" : "+v"(escaped) : : "memory");
  }
#else
  for (int i = threadIdx.x; i < HID * HID; i += 256) Bs[i] = Bp[i];
#endif
  __syncthreads();

  int wave  = threadIdx.x >> 5;
  int lane  = threadIdx.x & 31;
  int mtile = blockIdx.x * 8 + wave;
  if (mtile * 16 >= M) return;          // whole-wave uniform exit (EXEC stays full)

  int rlo   = lane & 15;                // M (for A) / N (for B,C,D) within tile
  int khalf = lane >> 4;                // selects K pair {0,1} vs {2,3}
  int row   = mtile * 16 + rlo;
  int rowc  = row < M ? row : M - 1;
  float msk = row < M ? 1.f : 0.f;      // mask, not branch: EXEC all-ones at WMMA
  const float* Arow = A + rowc * HID;

  v8f acc[8];
#pragma unroll
  for (int t = 0; t < 8; ++t) acc[t] = (v8f){0.f, 0.f, 0.f, 0.f, 0.f, 0.f, 0.f, 0.f};

  bool tail = (mtile * 16 + 16 > M);    // wave-uniform
  if (tail) kloop128<true>(Arow, Bs, msk, khalf, rlo, acc);
  else      kloop128<false>(Arow, Bs, msk, khalf, rlo, acc);

#pragma unroll
  for (int t = 0; t < 8; ++t) {
    int col = t * 16 + rlo;
    float bv = bias ? bias[col] : 0.f;
#pragma unroll
    for (int r = 0; r < 8; ++r) {
      int m = mtile * 16 + khalf * 8 + r;   // C/D layout: VGPR r -> M = r + 8*khalf
      if (m < M) {
        float v = acc[t][r] + bv;
        if (act) v = tanhf(v);
        C[m * HID + col] = v;
      }
    }
  }
}

// gsum[batch[n],:] += H[n,:] ; gcnt[batch[n]] += 1
__global__ void k_pool(const float* __restrict__ H, const int* __restrict__ batch,
                       float* gsum, float* gcnt, int N) {
  int i = blockIdx.x * blockDim.x + threadIdx.x;
  if (i >= N * HID) return;
  int n = i >> 7, f = i & 127;
  int g = batch[n];
  atomicAdd(&gsum[g * HID + f], H[i]);
  if (f == 0) atomicAdd(&gcnt[g], 1.0f);
}

__global__ void k_pooldiv(float* gsum, const float* gcnt, int G) {
  int i = blockIdx.x * blockDim.x + threadIdx.x;
  if (i >= G * HID) return;
  gsum[i] /= fmaxf(gcnt[i >> 7], 1.0f);
}

// out[g] = dot(Z[g,:], Wm2) + bm2 ; one wave32 per graph
__global__ void k_final(const float* __restrict__ Z, const float* __restrict__ Wm2,
                        const float* __restrict__ bm2, float* __restrict__ out, int G) {
  int g = blockIdx.x;
  int lane = threadIdx.x;
  float s = 0.f;
  for (int f = lane; f < HID; f += 32) s += Z[g * HID + f] * Wm2[f];
#pragma unroll
  for (int off = 16; off > 0; off >>= 1) s += __shfl_down(s, off, 32);
  if (lane == 0) out[g] = s + bm2[0];
}

extern "C" void kernel_launch(void* const* d_in, const int* in_sizes, int n_in,
                              void* d_out, int out_size, void* d_ws, size_t ws_size,
                              hipStream_t stream) {
  const float* x   = (const float*)d_in[0];
  const int*   ei  = (const int*)d_in[1];
  const int*   bat = (const int*)d_in[2];
  // d_in[3] = num_graphs scalar (device); grid dims need host value -> out_size
  const float* W1  = (const float*)d_in[4];
  const float* b1  = (const float*)d_in[5];
  const float* W2  = (const float*)d_in[6];
  const float* b2  = (const float*)d_in[7];
  const float* Wm1 = (const float*)d_in[8];
  const float* bm1 = (const float*)d_in[9];
  const float* Wm2 = (const float*)d_in[10];
  const float* bm2 = (const float*)d_in[11];

  int N = in_sizes[0] / 10;   // 100000
  int E = in_sizes[1] / 2;    // 1600000
  int G = out_size;           // 1000
  const int* src = ei;        // edge_index[0]
  const int* dst = ei + E;    // edge_index[1]

  // workspace carve-out (all f32):
  // BpW2[16K] | BpWm1[16K] | dinv[N] | T[N*128] | AGG[N*128] |
  // gsum[G*128] | gcnt[G] | Z[G*128]
  float* BpW2  = (float*)d_ws;
  float* BpWm1 = BpW2 + HID * HID;
  float* dinv  = BpWm1 + HID * HID;
  float* T     = dinv + N;
  float* AGG   = T + (size_t)N * HID;
  float* gsum  = AGG + (size_t)N * HID;
  float* gcnt  = gsum + (size_t)G * HID;
  float* Z     = gcnt + G;

  dim3 b256(256);
  int gN  = (N + 255) / 256;
  int gE  = (E + 255) / 256;
  int gNF = (N * HID + 255) / 256;
  int gGF = (G * HID + 255) / 256;
  int gW  = (HID * HID + 255) / 256;
  long long ef = (long long)E * HID;
  int gEF = (int)((ef + 255) / 256);
  int gemmN = (((N + 15) / 16) + 7) / 8;
  int gemmG = (((G + 15) / 16) + 7) / 8;

  // one-shot weight repacks for the WMMA GEMMs
  k_repack<<<gW, b256, 0, stream>>>(W2,  BpW2);
  k_repack<<<gW, b256, 0, stream>>>(Wm1, BpWm1);

  // degrees (self-loop => start at 1), then dinv = rsqrt(deg)
  k_fill1 <<<gN, b256, 0, stream>>>(dinv, N);
  k_degacc<<<gE, b256, 0, stream>>>(dst, dinv, E);
  k_rsqrt <<<gN, b256, 0, stream>>>(dinv, N);

  // layer 1: transform -> propagate -> bias+relu (h1 lands in T)
  k_xw1     <<<gNF, b256, 0, stream>>>(x, W1, T, N);
  k_selfinit<<<gNF, b256, 0, stream>>>(T, dinv, AGG, N);
  k_scatter <<<gEF, b256, 0, stream>>>(src, dst, dinv, T, AGG, E);
  k_biasrelu<<<gNF, b256, 0, stream>>>(AGG, b1, T, N);

  // layer 2: WMMA GEMM (t2 -> AGG), propagate into T, bias+relu (h2 -> AGG)
  k_gemm128_wmma<<<gemmN, b256, 0, stream>>>(T, BpW2, AGG, N, nullptr, 0);
  k_selfinit    <<<gNF,   b256, 0, stream>>>(AGG, dinv, T, N);
  k_scatter     <<<gEF,   b256, 0, stream>>>(src, dst, dinv, AGG, T, E);
  k_biasrelu    <<<gNF,   b256, 0, stream>>>(T, b2, AGG, N);

  // global mean pool (gsum,gcnt contiguous -> one zero pass)
  k_zero   <<<(G * HID + G + 255) / 256, b256, 0, stream>>>(gsum, G * HID + G);
  k_pool   <<<gNF, b256, 0, stream>>>(AGG, bat, gsum, gcnt, N);
  k_pooldiv<<<gGF, b256, 0, stream>>>(gsum, gcnt, G);

  // head: Z = tanh(g @ Wm1 + bm1) via WMMA, then per-graph dot with Wm2
  k_gemm128_wmma<<<gemmG, b256, 0, stream>>>(gsum, BpWm1, Z, G, bm1, 1);
  k_final<<<G, dim3(32), 0, stream>>>(Z, Wm2, bm2, (float*)d_out, G);
}